// GNNModel_50087908606622
// MI455X (gfx1250) — compile-verified
//
#include <hip/hip_runtime.h>
#include <hip/hip_bf16.h>

#define HID 128
#define OUTD 3

typedef float v2f __attribute__((ext_vector_type(2)));
typedef float v8f __attribute__((ext_vector_type(8)));

// ---------------- utility kernels ----------------

__global__ void k_zero(float* __restrict__ p, long long n) {
  long long t = (long long)blockIdx.x * blockDim.x + threadIdx.x;
  if (t < n) p[t] = 0.0f;
}

// deg starts at 1.0 (self-loop)
__global__ void k_deg_init(float* __restrict__ deg, int n) {
  int t = blockIdx.x * blockDim.x + threadIdx.x;
  if (t < n) deg[t] = 1.0f;
}

__global__ void k_deg_edges(const int* __restrict__ dst, float* __restrict__ deg, int E) {
  int t = blockIdx.x * blockDim.x + threadIdx.x;
  if (t < E) unsafeAtomicAdd(&deg[dst[t]], 1.0f);
}

__global__ void k_dinv(float* __restrict__ deg, int n) {
  int t = blockIdx.x * blockDim.x + threadIdx.x;
  if (t < n) deg[t] = rsqrtf(fmaxf(deg[t], 1.0f));
}

__global__ void k_norm(const int* __restrict__ src, const int* __restrict__ dst,
                       const float* __restrict__ dinv, float* __restrict__ nrm, int E) {
  int t = blockIdx.x * blockDim.x + threadIdx.x;
  if (t < E) nrm[t] = dinv[src[t]] * dinv[dst[t]];
}

// ---------------- fp32 WMMA GEMM: H[M,N] = X[M,K] @ W[K,N] ----------------
// One wave computes a 16x64 tile: 4 independent V_WMMA_F32_16X16X4_F32
// accumulator chains share one A fragment per K-step (A traffic /4, and the
// 4 chains hide the WMMA->WMMA accumulator RAW hazard).
// A 16x4 layout: lanes 0-15 rows 0-15 K={k,k+1}; lanes 16-31 same rows K={k+2,k+3}.
// B 4x16 mirrors A; C/D: vgpr v -> row v (lanes 0-15) / row v+8 (lanes 16-31).
// Requires: N % 64 == 0, M % 16 == 0, K % 4 == 0.
__global__ void k_gemm_wmma(const float* __restrict__ X, const float* __restrict__ W,
                            float* __restrict__ H, int M, int K, int N) {
  int wave = (int)((blockIdx.x * (long long)blockDim.x + threadIdx.x) >> 5);
  int lane = threadIdx.x & 31;
  int ntiles = N >> 6;                           // 64-wide N tiles
  int mtiles = M >> 4;
  if (wave >= mtiles * ntiles) return;           // wave-uniform exit (EXEC stays all-ones)
  int mt = wave / ntiles;
  int nt = wave % ntiles;

  int half = lane >> 4;                          // 0: K pair {k,k+1}; 1: {k+2,k+3}
  int l16  = lane & 15;

  const float* xrow = X + (long long)(mt * 16 + l16) * K;
  const float* wc   = W + nt * 64 + l16;

  v8f c0 = {}, c1 = {}, c2 = {}, c3 = {};
#pragma unroll 2
  for (int k0 = 0; k0 < K; k0 += 4) {
    int kk = k0 + 2 * half;
    v2f a; a.x = xrow[kk]; a.y = xrow[kk + 1];
    const float* w0 = wc + (long long)kk * N;    // row kk
    const float* w1 = w0 + N;                    // row kk+1
    v2f b0; b0.x = w0[0];  b0.y = w1[0];
    v2f b1; b1.x = w0[16]; b1.y = w1[16];
    v2f b2; b2.x = w0[32]; b2.y = w1[32];
    v2f b3; b3.x = w0[48]; b3.y = w1[48];
    c0 = __builtin_amdgcn_wmma_f32_16x16x4_f32(false, a, false, b0, (short)0, c0, false, false);
    c1 = __builtin_amdgcn_wmma_f32_16x16x4_f32(false, a, false, b1, (short)0, c1, false, false);
    c2 = __builtin_amdgcn_wmma_f32_16x16x4_f32(false, a, false, b2, (short)0, c2, false, false);
    c3 = __builtin_amdgcn_wmma_f32_16x16x4_f32(false, a, false, b3, (short)0, c3, false, false);
  }

  float* hrow = H + (long long)(mt * 16 + 8 * half) * N + nt * 64 + l16;
#pragma unroll
  for (int v = 0; v < 8; v++) {
    float* r = hrow + (long long)v * N;
    r[0]  = c0[v];
    r[16] = c1[v];
    r[32] = c2[v];
    r[48] = c3[v];
  }
}

// ---------------- edge scatter: acc[dst] += norm * H[src] ----------------
// 32 threads per edge, float4 (4 features) each -> 128 features.
__global__ void k_scatter(const int* __restrict__ src, const int* __restrict__ dst,
                          const float* __restrict__ nrm, const float* __restrict__ H,
                          float* __restrict__ acc, int E) {
  long long t = (long long)blockIdx.x * blockDim.x + threadIdx.x;
  int e = (int)(t >> 5);
  if (e >= E) return;
  int q = ((int)t & 31) << 2;
  int s = src[e], d = dst[e];
  float w = nrm[e];
  float4 h = *(const float4*)(H + (long long)s * HID + q);
  float* a = acc + (long long)d * HID + q;
  unsafeAtomicAdd(a + 0, w * h.x);
  unsafeAtomicAdd(a + 1, w * h.y);
  unsafeAtomicAdd(a + 2, w * h.z);
  unsafeAtomicAdd(a + 3, w * h.w);
}

// acc[i,f] = relu(acc[i,f] + dinv[i]^2 * H[i,f] + bias[f])   (self-loop folded in)
__global__ void k_finalize(float* __restrict__ acc, const float* __restrict__ H,
                           const float* __restrict__ dinv, const float* __restrict__ bias,
                           int Nn) {
  long long t = (long long)blockIdx.x * blockDim.x + threadIdx.x;
  if (t >= (long long)Nn * HID) return;
  int i = (int)(t >> 7);
  int f = (int)(t & (HID - 1));
  float di = dinv[i];
  float v = acc[t] + di * di * H[t] + bias[f];
  acc[t] = fmaxf(v, 0.0f);
}

// pooled sums/counts per graph
__global__ void k_pool(const float* __restrict__ h, const int* __restrict__ batch,
                       float* __restrict__ sums, float* __restrict__ cnt, int Nn) {
  long long t = (long long)blockIdx.x * blockDim.x + threadIdx.x;
  if (t >= (long long)Nn * HID) return;
  int i = (int)(t >> 7);
  int f = (int)(t & (HID - 1));
  int g = batch[i];
  unsafeAtomicAdd(&sums[(long long)g * HID + f], h[t]);
  if (f == 0) unsafeAtomicAdd(&cnt[g], 1.0f);
}

// out[g,o] = (sums[g]/max(cnt,1)) @ Wf + bf    (G*OUTD threads, one block)
__global__ void k_final(const float* __restrict__ sums, const float* __restrict__ cnt,
                        const float* __restrict__ Wf, const float* __restrict__ bf,
                        float* __restrict__ out, int G) {
  int t = blockIdx.x * blockDim.x + threadIdx.x;
  if (t >= G * OUTD) return;
  int g = t / OUTD, o = t % OUTD;
  float inv = 1.0f / fmaxf(cnt[g], 1.0f);
  float a = bf[o];
  const float* s = sums + (long long)g * HID;
#pragma unroll 4
  for (int f = 0; f < HID; f++) a += s[f] * inv * Wf[f * OUTD + o];
  out[t] = a;
}

// ---------------- launch ----------------

extern "C" void kernel_launch(void* const* d_in, const int* in_sizes, int n_in,
                              void* d_out, int out_size, void* d_ws, size_t ws_size,
                              hipStream_t stream) {
  const float* x   = (const float*)d_in[0];
  const int*   ei  = (const int*)d_in[1];
  const int*   bat = (const int*)d_in[2];
  const float* W1  = (const float*)d_in[3];
  const float* b1  = (const float*)d_in[4];
  const float* W2  = (const float*)d_in[5];
  const float* b2  = (const float*)d_in[6];
  const float* Wf  = (const float*)d_in[7];
  const float* bf  = (const float*)d_in[8];
  float* out = (float*)d_out;

  const int Nn = in_sizes[2];
  const int E  = in_sizes[1] / 2;
  const int IN = in_sizes[0] / Nn;
  const int G  = out_size / OUTD;
  const int* src = ei;
  const int* dst = ei + E;

  // workspace layout (floats): A | B | dinv | norm | sums | cnt
  float* A    = (float*)d_ws;
  float* B    = A    + (long long)Nn * HID;
  float* dinv = B    + (long long)Nn * HID;
  float* nrm  = dinv + Nn;
  float* sums = nrm  + E;
  float* cnt  = sums + (long long)G * HID;

  const int TB = 256;
  const long long nf = (long long)Nn * HID;
  const int nBlkN  = (Nn + TB - 1) / TB;
  const int nBlkE  = (E + TB - 1) / TB;
  const int nBlkF  = (int)((nf + TB - 1) / TB);
  const int nBlkSc = (int)(((long long)E * (HID / 4) + TB - 1) / TB);
  // GEMM: one wave per 16x64 tile
  const long long gthr = (long long)(Nn / 16) * (HID / 64) * 32;
  const int nBlkG = (int)((gthr + TB - 1) / TB);

  // degrees -> dinv -> per-edge norm
  k_deg_init <<<nBlkN, TB, 0, stream>>>(dinv, Nn);
  k_deg_edges<<<nBlkE, TB, 0, stream>>>(dst, dinv, E);
  k_dinv     <<<nBlkN, TB, 0, stream>>>(dinv, Nn);
  k_norm     <<<nBlkE, TB, 0, stream>>>(src, dst, dinv, nrm, E);

  // layer 1: B = X @ W1 ; A = scatter(B) ; A = relu(A + dinv^2*B + b1)
  k_gemm_wmma<<<nBlkG, TB, 0, stream>>>(x, W1, B, Nn, IN, HID);
  k_zero     <<<nBlkF, TB, 0, stream>>>(A, nf);
  k_scatter  <<<nBlkSc, TB, 0, stream>>>(src, dst, nrm, B, A, E);
  k_finalize <<<nBlkF, TB, 0, stream>>>(A, B, dinv, b1, Nn);

  // layer 2: B = A @ W2 ; A = scatter(B) ; A = relu(A + dinv^2*B + b2)
  k_gemm_wmma<<<nBlkG, TB, 0, stream>>>(A, W2, B, Nn, HID, HID);
  k_zero     <<<nBlkF, TB, 0, stream>>>(A, nf);    // GEMM2 reads A before this (stream order)
  k_scatter  <<<nBlkSc, TB, 0, stream>>>(src, dst, nrm, B, A, E);
  k_finalize <<<nBlkF, TB, 0, stream>>>(A, B, dinv, b2, Nn);

  // pooling + head
  k_zero <<<(int)(((long long)G * HID + G + TB - 1) / TB), TB, 0, stream>>>(sums, (long long)G * HID + G);
  k_pool <<<nBlkF, TB, 0, stream>>>(A, bat, sums, cnt, Nn);
  k_final<<<1, ((G * OUTD + 31) / 32) * 32, 0, stream>>>(sums, cnt, Wf, bf, out, G);
}